// MultiLoss_39462159515734
// MI455X (gfx1250) — compile-verified
//
#include <hip/hip_runtime.h>
#include <hip/hip_bf16.h>

typedef float v2f __attribute__((ext_vector_type(2)));
typedef float v8f __attribute__((ext_vector_type(8)));

// Full-precision f32 cross-lane sum of one value per lane using the matrix pipe.
// A (16x4 f32): lane m (0-15) supplies A[m][0]=s, lanes 16-31 supply A[m][2]=s,
// second VGPR (K=1/K=3) = 0. B (4x16) = all ones. Then D[m][n] = s_m + s_{m+16}.
// Lanes 0-15 hold rows 0-7 in d[0..7]; lanes 16-31 hold rows 8-15.
// Sum d[0..7] intra-lane, then one xor-16 shuffle -> every lane has the wave sum.
__device__ __forceinline__ float wave_sum_wmma(float s) {
    v2f a; a[0] = s;    a[1] = 0.0f;
    v2f b; b[0] = 1.0f; b[1] = 1.0f;
    v8f c = {};
    v8f d = __builtin_amdgcn_wmma_f32_16x16x4_f32(
        /*neg_a=*/false, a, /*neg_b=*/false, b,
        /*c_mod=*/(short)0, c, /*reuse_a=*/false, /*reuse_b=*/false);
    float t = ((d[0] + d[1]) + (d[2] + d[3])) + ((d[4] + d[5]) + (d[6] + d[7]));
    t += __shfl_xor(t, 16, 32);
    return t;
}

// loss = sum_{b,t} w[b] * log(pol[b,t,act[b,t]]),  w[b] = (ea[b]-adv[b])/B
__global__ void pg_loss_partial(const float* __restrict__ pol,
                                const int*   __restrict__ act,
                                const float* __restrict__ ea,
                                const float* __restrict__ adv,
                                float*       __restrict__ partials,
                                int nElem, int T, int A, float invB) {
    const int tid    = blockIdx.x * blockDim.x + threadIdx.x;
    const int stride = gridDim.x * blockDim.x;

    float s = 0.0f;
    for (int e = tid; e < nElem; e += stride) {
        const unsigned b = (unsigned)e / (unsigned)T;
        const int a = act[e];
        const float p = pol[(size_t)e * (size_t)A + (size_t)(unsigned)a];
        const float w = (ea[b] - adv[b]) * invB;
        s = fmaf(w, logf(p), s);
    }

    // Wave-level reduce via f32 WMMA (EXEC is all-1s here: uniform control flow).
    const float ws = wave_sum_wmma(s);

    __shared__ float lds[8];
    const int wid  = threadIdx.x >> 5;
    const int lane = threadIdx.x & 31;
    if (lane == 0) lds[wid] = ws;
    __syncthreads();

    if (wid == 0) {
        float v = (lane < (int)(blockDim.x >> 5)) ? lds[lane] : 0.0f;
        v += __shfl_xor(v, 1, 32);
        v += __shfl_xor(v, 2, 32);
        v += __shfl_xor(v, 4, 32);
        if (lane == 0) partials[blockIdx.x] = v;
    }
}

// Single-wave deterministic finalize: strided scalar accumulate + WMMA reduce.
__global__ void pg_loss_finalize(const float* __restrict__ partials,
                                 int n, float* __restrict__ out) {
    float s = 0.0f;
    for (int i = threadIdx.x; i < n; i += 32) s += partials[i];
    const float t = wave_sum_wmma(s);
    if (threadIdx.x == 0) out[0] = t;
}

extern "C" void kernel_launch(void* const* d_in, const int* in_sizes, int n_in,
                              void* d_out, int out_size, void* d_ws, size_t ws_size,
                              hipStream_t stream) {
    const float* pol = (const float*)d_in[0];   // [B, T, A] f32
    const int*   act = (const int*)d_in[1];     // [B, T]    int
    const float* ea  = (const float*)d_in[2];   // [B, 1]    f32
    const float* adv = (const float*)d_in[3];   // [B, 1]    f32

    const int nElem = in_sizes[1];              // B*T
    const int Bn    = in_sizes[2];              // B
    const int A     = in_sizes[0] / nElem;      // 64
    const int T     = nElem / Bn;               // 200
    const float invB = 1.0f / (float)Bn;

    float* partials = (float*)d_ws;

    const int threads = 256;                    // 8 waves per block (wave32)
    const int blocks  = 1024;                   // fixed -> deterministic tree

    pg_loss_partial<<<blocks, threads, 0, stream>>>(pol, act, ea, adv, partials,
                                                    nElem, T, A, invB);
    pg_loss_finalize<<<1, 32, 0, stream>>>(partials, blocks, (float*)d_out);
}